// GaussianStep_32667521253782
// MI455X (gfx1250) — compile-verified
//
#include <hip/hip_runtime.h>
#include <hip/hip_bf16.h>

// Batched Kalman step, B=131072, STATE=16, OBS=8, fp32.
// One wave32 per batch item; 16x16 (OBS padded to 16) matmuls via
// v_wmma_f32_16x16x4_f32 (inner-dim-8 products use only 2 WMMA);
// staging via global_load_async_to_lds_b128/b64 (ASYNCcnt);
// 8x8 SPD inverse via cooperative Gauss-Jordan in LDS.

#define KB      131072
#define STATE   16
#define OBS     8
#define WPB     4            // waves per block
#define THREADS (WPB * 32)
#define STRIDE  20           // LDS row stride in floats (conflict-free cols, 16B rows)
#define SLOT    (16 * STRIDE)
#define NSLOT   9
#define PERWAVE (NSLOT * SLOT + 64)

typedef float v2f __attribute__((ext_vector_type(2)));
typedef float v8f __attribute__((ext_vector_type(8)));

// LDS ops of a wave are processed in order at the LDS (ISA 7.3), so cross-lane
// store->load within one wave only needs compile-time ordering.
#define CBAR() asm volatile("" ::: "memory")

// Async global->LDS staging (GV mode). LDS address = low 32 bits of the
// generic pointer (flat->LDS truncation per ISA 10.2).
#define ASYNC_B128(ldsptr, gptr)                                               \
  asm volatile("global_load_async_to_lds_b128 %0, %1, off"                     \
               :: "v"((unsigned)(uintptr_t)(ldsptr)), "v"(gptr) : "memory")
#define ASYNC_B64(ldsptr, gptr)                                                \
  asm volatile("global_load_async_to_lds_b64 %0, %1, off"                      \
               :: "v"((unsigned)(uintptr_t)(ldsptr)), "v"(gptr) : "memory")
#define ASYNC_WAIT() asm volatile("s_wait_asynccnt 0" ::: "memory")

// ---- WMMA helpers (layouts per CDNA5 ISA 7.12.2, wave32) -------------------
// C/D 16x16 f32: lane l, vgpr r -> (M = r + 8*(l>=16), N = l&15)
// A 16x4  f32: lanes 0-15 M=0..15; vgpr0 = K {k0, k0+2}, vgpr1 = K {k0+1, k0+3}
// B 4x16  f32: vgpr0 = rows {k0, k0+2} striped over lane halves, vgpr1 = {k0+1, k0+3}

template <int KN, bool TA, bool TB>
__device__ __forceinline__ v8f mmN(const float* __restrict__ A,
                                   const float* __restrict__ Bm,
                                   v8f c, int lane) {
  const int r = lane & 15;                 // also N index for B
  const int khalf = (lane >> 4) << 1;      // +2 for upper lane half
#pragma unroll
  for (int k0 = 0; k0 < KN; k0 += 4) {
    const int kk = k0 + khalf;
    v2f a, b;
    if (TA) { a.x = A[kk * STRIDE + r];  a.y = A[(kk + 1) * STRIDE + r]; }
    else    { a.x = A[r * STRIDE + kk];  a.y = A[r * STRIDE + kk + 1];  }
    if (TB) { b.x = Bm[r * STRIDE + kk]; b.y = Bm[r * STRIDE + kk + 1]; }
    else    { b.x = Bm[kk * STRIDE + r]; b.y = Bm[(kk + 1) * STRIDE + r]; }
    c = __builtin_amdgcn_wmma_f32_16x16x4_f32(false, a, false, b,
                                              (short)0, c, false, false);
  }
  return c;
}

__device__ __forceinline__ void stC(float* m, v8f c, int lane) {
  const int n = lane & 15;
  const int mo = (lane >> 4) * 8;
#pragma unroll
  for (int rr = 0; rr < 8; ++rr) m[(mo + rr) * STRIDE + n] = c[rr];
}

__device__ __forceinline__ v8f ldCD(const float* m, int lane) {
  const int n = lane & 15;
  const int mo = (lane >> 4) * 8;
  v8f c;
#pragma unroll
  for (int rr = 0; rr < 8; ++rr) c[rr] = m[(mo + rr) * STRIDE + n];
  return c;
}

__global__ __launch_bounds__(THREADS)
void kalman_step_kernel(const float* __restrict__ inputg,
                        const float* __restrict__ meang,
                        const float* __restrict__ covg,
                        const float* __restrict__ Hg,
                        const float* __restrict__ Rg,
                        const float* __restrict__ Fg,
                        const float* __restrict__ Qg,
                        float* __restrict__ out) {
  __shared__ float lds[WPB * PERWAVE];

  const int wave = threadIdx.x >> 5;
  const int lane = threadIdx.x & 31;
  const int item = blockIdx.x * WPB + wave;

  float* base = lds + wave * PERWAVE;
  float* sCov = base + 0 * SLOT;   // cov, later new_cov
  float* sH   = base + 1 * SLOT;   // H padded, later T3 = K R
  float* sR   = base + 2 * SLOT;   // R padded
  float* sF   = base + 3 * SLOT;   // F
  float* sT1  = base + 4 * SLOT;   // T1 = cov H^T, later T2, T4
  float* sS   = base + 5 * SLOT;   // S / Gauss-Jordan augmented system
  float* sSi  = base + 6 * SLOT;   // S^-1 padded
  float* sK   = base + 7 * SLOT;   // K
  float* sP1  = base + 8 * SLOT;   // p1 = I - K H
  float* vecM = base + NSLOT * SLOT;       // mean[16]
  float* vecR = vecM + 16;                 // resid[8]
  float* vecN = vecR + 8;                  // new_mean[16]

  const int n  = lane & 15;
  const int mo = (lane >> 4) * 8;

  // ---- async-stage cov/F/H/R directly into LDS ----------------------------
  {
    const float* cg = covg + (size_t)item * 256;
    const float* fg = Fg + (size_t)item * 256;
    const float* hg = Hg + (size_t)item * 128;
    const float* rg = Rg + (size_t)item * 64;
    const int rowq = lane >> 2;            // 0..7
    const int col4 = (lane & 3) * 4;       // {0,4,8,12}
    ASYNC_B128(sCov + rowq * STRIDE + col4,       cg + lane * 4);
    ASYNC_B128(sCov + (8 + rowq) * STRIDE + col4, cg + 128 + lane * 4);
    ASYNC_B128(sF + rowq * STRIDE + col4,         fg + lane * 4);
    ASYNC_B128(sF + (8 + rowq) * STRIDE + col4,   fg + 128 + lane * 4);
    ASYNC_B128(sH + rowq * STRIDE + col4,         hg + lane * 4);
    ASYNC_B64 (sR + rowq * STRIDE + (lane & 3) * 2, rg + lane * 2);
  }
  // zero-pad regions DISJOINT from async-written entries (no DS/ASYNC races)
#pragma unroll
  for (int j = 0; j < 8; ++j) {
    int i = lane + 32 * j;                 // 0..255
    int r = i >> 4, c = i & 15;
    if (r >= 8) sH[r * STRIDE + c] = 0.0f;
    if (r >= 8 || c >= 8) sR[r * STRIDE + c] = 0.0f;
  }
  if (lane < 16) vecM[lane] = meang[(size_t)item * 16 + lane];
  ASYNC_WAIT();
  CBAR();

  // ---- residual: resid = input - H mean  (lanes 0..7) ---------------------
  if (lane < OBS) {
    float s = 0.0f;
#pragma unroll
    for (int k = 0; k < 16; ++k) s += sH[lane * STRIDE + k] * vecM[k];
    vecR[lane] = inputg[(size_t)item * OBS + lane] - s;
  }
  CBAR();

  const v8f zf = {};

  // ---- T1 = cov @ H^T  (inner 16) -----------------------------------------
  v8f t1 = mmN<16, false, true>(sCov, sH, zf, lane);
  stC(sT1, t1, lane);
  CBAR();

  // ---- S = H @ T1 + R  (inner 16) -----------------------------------------
  v8f sacc = mmN<16, false, false>(sH, sT1, ldCD(sR, lane), lane);
  stC(sS, sacc, lane);
  CBAR();

  // ---- Gauss-Jordan inverse of 8x8 S (SPD, no pivoting) -------------------
  {
    const int gr  = lane >> 2;             // row 0..7
    const int gc0 = (lane & 3) * 4;        // cols {0,4,8,12}
#pragma unroll
    for (int j = 0; j < 4; ++j) {          // augment with identity in cols 8..15
      int c = gc0 + j;
      if (c >= 8) sS[gr * STRIDE + c] = (c - 8 == gr) ? 1.0f : 0.0f;
    }
    CBAR();
    for (int p = 0; p < 8; ++p) {
      float invp = 1.0f / sS[p * STRIDE + p];
      float f = sS[gr * STRIDE + p];
      CBAR();
      if (gr == p) {
#pragma unroll
        for (int j = 0; j < 4; ++j) sS[p * STRIDE + gc0 + j] *= invp;
      }
      CBAR();
      if (gr != p) {
#pragma unroll
        for (int j = 0; j < 4; ++j)
          sS[gr * STRIDE + gc0 + j] -= f * sS[p * STRIDE + gc0 + j];
      }
      CBAR();
    }
#pragma unroll
    for (int j = 0; j < 8; ++j) {          // padded S^-1 into sSi
      int i = lane + 32 * j;
      int r = i >> 4, c = i & 15;
      float v = (r < 8 && c < 8) ? sS[r * STRIDE + 8 + c] : 0.0f;
      sSi[r * STRIDE + c] = v;
    }
  }
  CBAR();

  // ---- K = T1 @ S^-1  (inner 8: T1 cols 8..15 == 0) -----------------------
  v8f kacc = mmN<8, false, false>(sT1, sSi, zf, lane);
  stC(sK, kacc, lane);
  CBAR();

  // ---- new_mean = mean + K resid  (lanes 0..15) ---------------------------
  if (lane < 16) {
    float nm = vecM[lane];
#pragma unroll
    for (int o = 0; o < OBS; ++o) nm += sK[lane * STRIDE + o] * vecR[o];
    vecN[lane] = nm;
  }
  CBAR();

  // ---- p1 = I - K @ H  (inner 8) ------------------------------------------
  v8f kh = mmN<8, false, false>(sK, sH, zf, lane);
  v8f p1;
#pragma unroll
  for (int rr = 0; rr < 8; ++rr)
    p1[rr] = (((mo + rr) == n) ? 1.0f : 0.0f) - kh[rr];
  stC(sP1, p1, lane);
  CBAR();

  // ---- new_cov = p1 cov p1^T + K R K^T  (Joseph form) ---------------------
  v8f t2 = mmN<16, false, false>(sP1, sCov, zf, lane);  // T2 = p1 cov
  stC(sT1, t2, lane);
  CBAR();
  v8f nc = mmN<16, false, true>(sT1, sP1, zf, lane);    // T2 p1^T
  v8f t3 = mmN<8, false, false>(sK, sR, zf, lane);      // T3 = K R (inner 8)
  stC(sH, t3, lane);                                    // H slot free now
  CBAR();
  nc = mmN<8, false, true>(sH, sK, nc, lane);           // + T3 K^T (inner 8)
  stC(sCov, nc, lane);                                  // cov slot <- new_cov
  CBAR();

  // ---- pred_mean = F new_mean  (lanes 0..15) ------------------------------
  if (lane < 16) {
    float pm = 0.0f;
#pragma unroll
    for (int k = 0; k < 16; ++k) pm += sF[lane * STRIDE + k] * vecN[k];
    out[(size_t)item * 16 + lane] = pm;
  }

  // ---- pred_cov = F new_cov F^T + Q ---------------------------------------
  v8f t4 = mmN<16, false, false>(sF, sCov, zf, lane);   // T4 = F new_cov
  stC(sT1, t4, lane);
  CBAR();
  v8f cq;
#pragma unroll
  for (int rr = 0; rr < 8; ++rr)
    cq[rr] = Qg[(size_t)item * 256 + (mo + rr) * 16 + n];
  v8f pc = mmN<16, false, true>(sT1, sF, cq, lane);     // T4 F^T + Q

  float* outc = out + (size_t)KB * 16 + (size_t)item * 256;
#pragma unroll
  for (int rr = 0; rr < 8; ++rr)
    outc[(mo + rr) * 16 + n] = pc[rr];
}

extern "C" void kernel_launch(void* const* d_in, const int* in_sizes, int n_in,
                              void* d_out, int out_size, void* d_ws, size_t ws_size,
                              hipStream_t stream) {
  (void)in_sizes; (void)n_in; (void)d_ws; (void)ws_size; (void)out_size;
  const float* inp  = (const float*)d_in[0];
  const float* mean = (const float*)d_in[1];
  const float* cov  = (const float*)d_in[2];
  const float* H    = (const float*)d_in[3];
  const float* R    = (const float*)d_in[4];
  const float* F    = (const float*)d_in[5];
  const float* Q    = (const float*)d_in[6];
  float* out = (float*)d_out;

  dim3 grid(KB / WPB);
  dim3 block(THREADS);
  kalman_step_kernel<<<grid, block, 0, stream>>>(inp, mean, cov, H, R, F, Q, out);
}